// DiagonalLinear_75015898792222
// MI455X (gfx1250) — compile-verified
//
#include <hip/hip_runtime.h>
#include <stdint.h>

// DiagonalLinear: out[b, i, j] = in[b, i, j] * diag[j]
// Pure bandwidth problem: 537 MB traffic -> ~23 us floor at 23.3 TB/s.
// CDNA5 path: async global->LDS stage for diag (ASYNCcnt), NT b128 streams
// for the big tensor.

typedef __attribute__((ext_vector_type(4))) float v4f;

#define COLS 4096
#define TPB 256
#define V4_PER_ROW (COLS / 4)          // 1024 float4 per row
#define V4_PER_THREAD (V4_PER_ROW / TPB) // 4 float4 per thread per row
#define ROWS_PER_BLOCK 8

__global__ __launch_bounds__(TPB) void diag_scale_kernel(
    const float* __restrict__ in,
    const float* __restrict__ diag,
    float* __restrict__ out,
    int total_rows)
{
    __shared__ __align__(16) float sdiag[COLS];

    const int tid = threadIdx.x;

    // ---- Stage diag (16 KiB) into LDS with CDNA5 async copy ----
    // Each thread issues 4 x b128 async loads; lane-linear so the wave's
    // addresses are contiguous 512B chunks. LDS byte address = low 32 bits
    // of the generic pointer (ISA 10.2).
    {
        uint32_t lds_base = (uint32_t)(uintptr_t)(void*)sdiag;
#pragma unroll
        for (int k = 0; k < 4; ++k) {
            uint32_t off = (uint32_t)((k * TPB + tid) * 16); // bytes
            asm volatile(
                "global_load_async_to_lds_b128 %0, %1, %2 offset:0"
                :
                : "v"(lds_base + off), "v"(off), "s"(diag)
                : "memory");
        }
#if __has_builtin(__builtin_amdgcn_s_wait_asynccnt)
        __builtin_amdgcn_s_wait_asynccnt(0);
#else
        asm volatile("s_wait_asynccnt 0" ::: "memory");
#endif
    }
    __syncthreads();

    // Hoist this thread's diag slice (16 floats) into VGPRs once; reuse
    // across all rows this block processes.
    const v4f* sdiag4 = (const v4f*)sdiag;
    v4f d[V4_PER_THREAD];
#pragma unroll
    for (int k = 0; k < V4_PER_THREAD; ++k)
        d[k] = sdiag4[tid + k * TPB];

    const v4f* __restrict__ in4 = (const v4f*)in;
    v4f* __restrict__ out4 = (v4f*)out;

    const long row0 = (long)blockIdx.x * ROWS_PER_BLOCK;

#pragma unroll 1
    for (int r = 0; r < ROWS_PER_BLOCK; ++r) {
        long row = row0 + r;
        if (row >= total_rows) break;
        long base = row * (long)V4_PER_ROW;
#pragma unroll
        for (int k = 0; k < V4_PER_THREAD; ++k) {
            long idx = base + tid + k * TPB;
            // Streaming 256 MiB in / 256 MiB out: non-temporal so we don't
            // churn L2 (dataset > 192 MB L2 anyway).
            v4f x = __builtin_nontemporal_load(in4 + idx);
            v4f y = x * d[k];
            __builtin_nontemporal_store(y, out4 + idx);
        }
    }
}

extern "C" void kernel_launch(void* const* d_in, const int* in_sizes, int n_in,
                              void* d_out, int out_size, void* d_ws, size_t ws_size,
                              hipStream_t stream)
{
    const float* in = (const float*)d_in[0];   // (4, 4096, 4096) f32
    const float* diag = (const float*)d_in[1]; // (4096,) f32
    float* out = (float*)d_out;

    int total_rows = out_size / COLS;          // 4 * 4096 = 16384
    int blocks = (total_rows + ROWS_PER_BLOCK - 1) / ROWS_PER_BLOCK; // 2048

    diag_scale_kernel<<<blocks, TPB, 0, stream>>>(in, diag, out, total_rows);
}